// Blur_64776696758592
// MI455X (gfx1250) — compile-verified
//
#include <hip/hip_runtime.h>
#include <math.h>

// Problem geometry (from reference)
#define NB   4
#define DZ   32
#define HY   192
#define WX   192
#define KZ   9
#define KXY  15
#define ZP   4
#define XYP  7

// PSF in workspace: per-batch rows padded to 16 floats ([15] = 0)
#define PSF_ROW        16
#define PSF_PER_B      (KZ * KXY * PSF_ROW)   // 2160 floats
#define PSF_TOTAL_OFF  (NB * PSF_PER_B)       // 8640: global sum lives here

typedef __attribute__((ext_vector_type(2))) float v2f;
typedef __attribute__((ext_vector_type(4))) float v4f;
typedef __attribute__((ext_vector_type(8))) float v8f;

typedef __attribute__((address_space(1))) int gas_int;   // global
typedef __attribute__((address_space(3))) int las_int;   // LDS

// --------------------------------------------------------------------------
// CDNA5 async global->LDS copy (ASYNCcnt path), with safe fallbacks.
// Builtin signature (probe-confirmed): (int AS1*, int AS3*, Imm i32, Imm i32)
// --------------------------------------------------------------------------
__device__ __forceinline__ void async_g2l_b32(const float* gp, float* lp) {
#if __has_builtin(__builtin_amdgcn_global_load_async_to_lds_b32)
  __builtin_amdgcn_global_load_async_to_lds_b32(
      (gas_int*)gp, (las_int*)lp, 0, 0);
#else
  *lp = *gp;   // synchronous fallback (still correct)
#endif
}

__device__ __forceinline__ void wait_async0() {
#if __has_builtin(__builtin_amdgcn_s_wait_asynccnt)
  __builtin_amdgcn_s_wait_asynccnt(0);
#else
  asm volatile("s_wait_asynccnt 0x0" ::: "memory");
#endif
}

// ---------------------------------------------------------------------------
// Kernel 1: generate un-normalized PSF + global sum (single block).
// psf[b,kz,kh,kw] = 1 - exp(-alpha_b * lat_b(kh,kw) * ax_b(kz))
// ---------------------------------------------------------------------------
__global__ __launch_bounds__(256)
void psf_gen(const float* __restrict__ bet_xy, const float* __restrict__ bet_z,
             const float* __restrict__ alpha, float* __restrict__ ws) {
  __shared__ float red[256];
  const int tid = threadIdx.x;
  const float PI = 3.14159265358979323846f;
  float local = 0.f;
  for (int idx = tid; idx < NB * KZ * KXY * KXY; idx += 256) {
    int b  = idx / (KZ * KXY * KXY);
    int r  = idx % (KZ * KXY * KXY);
    int kz = r / (KXY * KXY);
    int r2 = r % (KXY * KXY);
    int kh = r2 / KXY;
    int kw = r2 % KXY;
    float bxy = bet_xy[b], bz = bet_z[b], al = alpha[b];
    float yd = fabsf((float)kh - 7.0f);
    float xd = fabsf((float)kw - 7.0f);
    float zd = fabsf((float)kz - 4.0f);
    float lat = expf(-(yd * yd + xd * xd) / (2.f * bxy * bxy)) / (2.f * PI * bxy * bxy);
    float ax  = expf(-(zd * zd) / (2.f * bz * bz)) / (sqrtf(2.f * PI) * bz);
    float v   = 1.f - expf(-al * lat * ax);
    ws[((b * KZ + kz) * KXY + kh) * PSF_ROW + kw] = v;
    local += v;
  }
  for (int i = tid; i < NB * KZ * KXY; i += 256) ws[i * PSF_ROW + 15] = 0.f;

  red[tid] = local;
  __syncthreads();
  for (int s = 128; s > 0; s >>= 1) {
    if (tid < s) red[tid] += red[tid + s];
    __syncthreads();
  }
  if (tid == 0) ws[PSF_TOTAL_OFF] = red[0];
}

// ---------------------------------------------------------------------------
// Kernel 2: depthwise 3D conv as Toeplitz implicit GEMM on V_WMMA_F32_16X16X4.
// Block = (b, z-pair, y-tile, x-quad): 256 threads = 8 waves.
//   wave = zoff*4 + xt : zoff in {0,1} picks z-plane, xt in {0..3} picks x-tile.
// Slab staged via GLOBAL_LOAD_ASYNC_TO_LDS (halo lanes write LDS zeros).
// ---------------------------------------------------------------------------
#define SLAB_Z 10          // 2 z outputs + 8 halo
#define SLAB_H 30          // 16 y outputs + 14 halo
#define SLAB_W 80          // 64 x outputs + 14 halo, padded to 80 (even stride)
#define XBLK   64          // x outputs per block (4 wave-tiles)
#define ZBLK   2           // z outputs per block
#define PADROW 48          // zero-padded Toeplitz row: taps at [16..30]
#define CONV_THREADS 256

__global__ __launch_bounds__(CONV_THREADS)
void conv_wmma(const float* __restrict__ x, const float* __restrict__ ws,
               float* __restrict__ out) {
  __shared__ float slab[SLAB_Z * SLAB_H * SLAB_W];   // 24000 f = 96000 B
  __shared__ float wpad[KZ * KXY * PADROW];          //  6480 f = 25920 B

  const int tid = threadIdx.x;
  const int bid = blockIdx.x;
  const int xb = bid % (WX / XBLK);                          // 3
  const int yt = (bid / (WX / XBLK)) % (HY / 16);            // 12
  const int zp = (bid / ((WX / XBLK) * (HY / 16))) % (DZ / ZBLK);
  const int b  = bid / ((WX / XBLK) * (HY / 16) * (DZ / ZBLK));
  const int z0 = zp * ZBLK;

  const int wave = tid >> 5;
  const int lane = tid & 31;

  // ---- Stage input slab via async global->LDS; halo -> LDS zeros. ----
  const int x0 = xb * XBLK - XYP;
  const int y0 = yt * 16 - XYP;
  for (int row = wave; row < SLAB_Z * SLAB_H; row += 8) {
    const int lz = row / SLAB_H;
    const int ly = row - lz * SLAB_H;
    const int zi = z0 - ZP + lz;
    const int yi = y0 + ly;
    const bool zy_ok = ((unsigned)zi < (unsigned)DZ) && ((unsigned)yi < (unsigned)HY);
    const float* grow = x + ((size_t)(b * DZ + zi) * HY + yi) * WX;
    float* lrow = &slab[row * SLAB_W];
    for (int j = lane; j < SLAB_W; j += 32) {
      const int xi = x0 + j;
      if (zy_ok && (unsigned)xi < (unsigned)WX) {
        async_g2l_b32(grow + xi, lrow + j);
      } else {
        lrow[j] = 0.f;                 // ds_store zero for halo
      }
    }
  }

  // ---- Stage zero-padded Toeplitz PSF rows (overlaps with async loads). ----
  for (int i = tid; i < KZ * KXY * PADROW; i += CONV_THREADS) {
    int r = i / PADROW;
    int t = (i % PADROW) - 16;
    float v = 0.f;
    if ((unsigned)t < 15u) v = ws[b * PSF_PER_B + r * PSF_ROW + t];
    wpad[i] = v;
  }

  wait_async0();        // drain ASYNCcnt before making LDS visible
  __syncthreads();

  const float inv = 1.0f / ws[PSF_TOTAL_OFF];

  const int xt   = wave & 3;    // x-tile within block
  const int zoff = wave >> 2;   // z-plane within block
  const int m    = lane & 15;   // M row for A operand == N col for B operand
  const int hi   = lane >> 4;   // half-wave: K = 2*hi + vgpr

  v8f acc = {0.f, 0.f, 0.f, 0.f, 0.f, 0.f, 0.f, 0.f};

  const int apad = 16 + hi * 2 - m;   // per-lane base into padded row, in [1,18]

  for (int kz = 0; kz < KZ; ++kz) {
    for (int ky = 0; ky < KXY; ++ky) {
      // B[j,n] = slab[zoff+kz, n+ky, xt*16 + j]; lane j = c*4 + hi*2 + {0,1}
      const float* brow =
          &slab[((zoff + kz) * SLAB_H + (m + ky)) * SLAB_W + xt * 16 + hi * 2];
      // A[m,j] = wpad[row, 16 + (j - m)]; always in-bounds, zeros pad the band.
      const float* arow = &wpad[(kz * KXY + ky) * PADROW + apad];
#pragma unroll
      for (int c = 0; c < 8; ++c) {
        const v2f bv = *(const v2f*)(brow + c * 4);   // 8B-aligned ds_load_b64
        v2f av = {arow[c * 4], arow[c * 4 + 1]};      // constant-offset ds loads
        acc = __builtin_amdgcn_wmma_f32_16x16x4_f32(
            false, av, false, bv, (short)0, acc, false, false);
      }
    }
  }

  // Store: lane = N = y; VGPR i = M = x (i + 8*hi) -> 8 contiguous x per lane.
  const int xo = xb * XBLK + xt * 16 + hi * 8;
  const int yo = yt * 16 + m;
  const int zo = z0 + zoff;
  float* o = out + ((size_t)(b * DZ + zo) * HY + yo) * WX + xo;
  v4f s0 = {acc[0] * inv, acc[1] * inv, acc[2] * inv, acc[3] * inv};
  v4f s1 = {acc[4] * inv, acc[5] * inv, acc[6] * inv, acc[7] * inv};
  *(v4f*)o = s0;          // 16B aligned: xo % 8 == 0, row stride 192 % 4 == 0
  *(v4f*)(o + 4) = s1;
}

// ---------------------------------------------------------------------------
extern "C" void kernel_launch(void* const* d_in, const int* in_sizes, int n_in,
                              void* d_out, int out_size, void* d_ws, size_t ws_size,
                              hipStream_t stream) {
  const float* x      = (const float*)d_in[0];
  const float* bet_xy = (const float*)d_in[1];
  const float* bet_z  = (const float*)d_in[2];
  const float* alpha  = (const float*)d_in[3];
  float* out = (float*)d_out;
  float* ws  = (float*)d_ws;

  psf_gen<<<1, 256, 0, stream>>>(bet_xy, bet_z, alpha, ws);

  const int nblocks = NB * (DZ / ZBLK) * (HY / 16) * (WX / XBLK);  // 2304
  conv_wmma<<<nblocks, CONV_THREADS, 0, stream>>>(x, ws, out);
}